// EdgeSIMPLEBatched_19791209300206
// MI455X (gfx1250) — compile-verified
//
#include <hip/hip_runtime.h>
#include <stdint.h>

// ---------------------------------------------------------------------------
// SIMPLE top-k subset sampling (EdgeSIMPLEBatched.forward) for MI455X/gfx1250.
// B = 2048 rows, N = 1024 (1000 real + pad), k = 32, T = 1.
// One wave32 per row: lane j carries ESP order j (lane 31 also carries j=32).
// Scores row is staged into LDS with the Tensor Data Mover (tensor_load_to_lds);
// suffix ESPs are checkpointed every 128 positions and recomputed blockwise in
// LDS (~30 KB/workgroup out of the WGP's 320 KB).
// Sampling scan is kept latency-lean: log-domain acceptance test (no exp on
// the loop-carried chain) and early-issued DS loads for the next step.
// ---------------------------------------------------------------------------

#define ENS   1000
#define NPAD  1024
#define KK    32
#define CBLK  128
#define NCKPT 9              // NPAD/CBLK + 1
#define NEGC  (-1.0e9f)
#define PADV  (-1.0e10f)

typedef __attribute__((ext_vector_type(4))) unsigned int uint32x4;
typedef __attribute__((ext_vector_type(4))) int          int32x4;
typedef __attribute__((ext_vector_type(8))) int          int32x8;

struct __align__(16) Smem {
  float flat[NPAD];             // logsigmoid(scores), padded
  float uarr[NPAD];             // unnormalized log-marginal per position
  float bits[NPAD];             // sampled inclusion bits
  float sbuf[CBLK + 1][KK + 1]; // suffix ESP rows for current block
  float cp[NCKPT][KK + 1];      // suffix ESP checkpoints
};

__device__ __forceinline__ float lae(float x, float y) {
  // logaddexp in f32; __expf underflows to 0 for very negative d, giving max().
  float m = fmaxf(x, y);
  float d = fminf(x, y) - m;
  return m + __logf(1.0f + __expf(d));
}

__device__ __forceinline__ float logsig(float x) {
  // -softplus(-x) + 1e-7
  return -(fmaxf(-x, 0.0f) + __logf(1.0f + __expf(-fabsf(x)))) + 1e-7f;
}

__device__ __forceinline__ float urand(uint32_t i, uint32_t row) {
  uint32_t h = i * 0x9E3779B9u ^ (row * 0x85EBCA6Bu + 0xC2B2AE35u);
  h ^= h >> 16; h *= 0x7FEB352Du;
  h ^= h >> 15; h *= 0x846CA68Bu;
  h ^= h >> 16;
  return (float)(h >> 8) * (1.0f / 16777216.0f);
}

__device__ __forceinline__ int clampr(int r) {
  return (r < 1) ? 1 : ((r > KK) ? KK : r);
}

__global__ __launch_bounds__(32) void simple_topk_kernel(
    const float* __restrict__ scores, float* __restrict__ out, int rows) {
  __shared__ Smem sm;
  const int row  = blockIdx.x;
  const int lane = threadIdx.x;
  if (row >= rows) return;

  // ---- Stage scores row into LDS via the Tensor Data Mover -----------------
#if __has_builtin(__builtin_amdgcn_tensor_load_to_lds)
  {
    uint64_t g = (uint64_t)(uintptr_t)(scores + (size_t)row * ENS);
    uint32_t ldsoff = (uint32_t)(uintptr_t)&sm.flat[0];   // generic low 32b = LDS offset
    // D# group0: count=1 | lds_addr | global_addr[56:0] | type=2
    uint32x4 g0 = { 1u, ldsoff, (uint32_t)g,
                    ((uint32_t)(g >> 32) & 0x1FFFFFFu) | 0x80000000u };
    // D# group1: data_size=4B; tensor_dim0=1000, tensor_dim1=1;
    //            tile_dim0=1000, tile_dim1=1; dim0_stride=1000
    int32x8 g1 = { (int)(2u << 16),
                   (int)((uint32_t)ENS << 16),
                   (int)(1u << 16),
                   (int)((uint32_t)ENS << 16),
                   1, ENS, 0, 0 };
    int32x4 z4 = { 0, 0, 0, 0 };
#if defined(__clang_major__) && __clang_major__ >= 23
    int32x8 z8 = { 0, 0, 0, 0, 0, 0, 0, 0 };
    __builtin_amdgcn_tensor_load_to_lds(g0, g1, z4, z4, z8, 0);
#else
    __builtin_amdgcn_tensor_load_to_lds(g0, g1, z4, z4, 0);
#endif
#if __has_builtin(__builtin_amdgcn_s_wait_tensorcnt)
    __builtin_amdgcn_s_wait_tensorcnt(0);
#else
    asm volatile("s_wait_tensorcnt 0x0" ::: "memory");
#endif
    asm volatile("" ::: "memory");
  }
#else
  for (int i = lane; i < ENS; i += 32) sm.flat[i] = scores[(size_t)row * ENS + i];
#endif

  // ---- logsigmoid + padding (in place) ------------------------------------
  for (int i0 = 0; i0 < NPAD; i0 += 32) {
    int i = i0 + lane;
    float v = (i < ENS) ? logsig(sm.flat[i]) : PADV;
    sm.flat[i] = v;
  }

  // ---- Phase 1: backward suffix-ESP scan, checkpoint every CBLK ----------
  // Lane j holds S[i][j]; Shi (uniform across lanes) holds S[i][32].
  float S   = (lane == 0) ? 0.0f : NEGC;
  float Shi = NEGC;
  sm.cp[NCKPT - 1][lane] = S;
  if (lane == 31) sm.cp[NCKPT - 1][KK] = Shi;
  for (int i = NPAD - 1; i >= 0; --i) {
    float a   = sm.flat[i];
    float sh  = __shfl_up(S, 1, 32);
    if (lane == 0) sh = NEGC;
    float s31 = __shfl(S, 31, 32);
    float nS  = lae(S, a + sh);
    Shi = lae(Shi, a + s31);
    S = nS;
    if ((i & (CBLK - 1)) == 0) {
      int ci = i >> 7;
      sm.cp[ci][lane] = S;
      if (lane == 31) sm.cp[ci][KK] = Shi;
    }
  }

  // ---- Phase 2: forward pass with blockwise suffix recomputation ----------
  float F   = (lane == 0) ? 0.0f : NEGC;
  float Fhi = NEGC;        // F[i][32], uniform across lanes
  int   r   = KK;          // remaining picks (wave-uniform)
  for (int b0 = 0; b0 < NPAD; b0 += CBLK) {
    // Rebuild S rows for i in [b0, b0+CBLK] from checkpoint S[b0+CBLK].
    int   ct  = (b0 >> 7) + 1;
    float Sb  = sm.cp[ct][lane];
    float Sbh = sm.cp[ct][KK];
    sm.sbuf[CBLK][lane] = Sb;
    if (lane == 31) sm.sbuf[CBLK][KK] = Sbh;
    for (int i = b0 + CBLK - 1; i >= b0; --i) {
      float a   = sm.flat[i];
      float sh  = __shfl_up(Sb, 1, 32);
      if (lane == 0) sh = NEGC;
      float s31 = __shfl(Sb, 31, 32);
      float nS  = lae(Sb, a + sh);
      Sbh = lae(Sbh, a + s31);
      Sb  = nS;
      int rr = i - b0;
      sm.sbuf[rr][lane] = Sb;
      if (lane == 31) sm.sbuf[rr][KK] = Sbh;
    }
    // Forward over the block: marginals + sampling + F recurrence.
    // Preload the sampling operands for the first position of the block.
    int   rc = clampr(r);
    float sn = sm.sbuf[1][rc - 1];   // S[i+1][rc-1]
    float si = sm.sbuf[0][rc];       // S[i][rc]
    for (int i = b0; i < b0 + CBLK; ++i) {
      int   rr = i - b0;
      float a  = sm.flat[i];
      // Sampling acceptance in log domain (no exp on the carried chain):
      //   u < clip(exp(a + sn - si),0,1)  <=>  log u < a + sn - si   (u in [0,1))
      float lu  = __logf(urand((uint32_t)i, (uint32_t)row) + 1e-12f);
      int   inc = ((r > 0) && (lu < a + sn - si)) ? 1 : 0;
      if (lane == 0) sm.bits[i] = (float)inc;
      r -= inc;
      // Issue next step's DS loads immediately; the lse reduction and the
      // F recurrence below cover their latency.
      if (rr + 1 < CBLK) {
        int rc2 = clampr(r);
        sn = sm.sbuf[rr + 2][rc2 - 1];
        si = sm.sbuf[rr + 1][rc2];
      }
      // Marginal: lse over j of F[i][j] + S[i+1][k-1-j]
      float t  = F + sm.sbuf[rr + 1][31 - lane];
      float mx = t;
      for (int m = 16; m; m >>= 1) mx = fmaxf(mx, __shfl_xor(mx, m, 32));
      float e = __expf(t - mx);
      for (int m = 16; m; m >>= 1) e += __shfl_xor(e, m, 32);
      float lse = mx + __logf(e);
      if (lane == 0) sm.uarr[i] = a + lse;
      // F recurrence
      float sh  = __shfl_up(F, 1, 32);
      if (lane == 0) sh = NEGC;
      float f31 = __shfl(F, 31, 32);
      float nF  = lae(F, a + sh);
      Fhi = lae(Fhi, a + f31);
      F = nF;
    }
  }
  float logZ = Fhi;   // F[N][k], already uniform across lanes

  // ---- Phase 3: normalize marginals, straight-through mask, store ---------
  const size_t base     = (size_t)row * ENS;
  const size_t margBase = (size_t)rows * ENS;
  for (int i0 = 0; i0 < ENS; i0 += 32) {
    int i = i0 + lane;
    if (i < ENS) {
      float m  = __expf(sm.uarr[i] - logZ);
      float b  = sm.bits[i];
      float mk = (b - m) + m;                 // stop_grad(s - m) + m forward value
      out[base + i]            = mk;
      out[margBase + base + i] = m;
    }
  }
}

extern "C" void kernel_launch(void* const* d_in, const int* in_sizes, int n_in,
                              void* d_out, int out_size, void* d_ws, size_t ws_size,
                              hipStream_t stream) {
  (void)n_in; (void)out_size; (void)d_ws; (void)ws_size;
  const float* scores = (const float*)d_in[0];
  float* out = (float*)d_out;
  int rows = in_sizes[0] / ENS;   // 8*256 = 2048
  if (rows <= 0) return;
  hipLaunchKernelGGL(simple_topk_kernel, dim3(rows), dim3(32), 0, stream,
                     scores, out, rows);
}